// GCN_74174085202052
// MI455X (gfx1250) — compile-verified
//
#include <hip/hip_runtime.h>

#define DD 128  // feature dim, fixed by the reference

typedef __attribute__((ext_vector_type(2))) float v2f;
typedef __attribute__((ext_vector_type(8))) float v8f;

// ---------------------------------------------------------------------------
// Degree / normalization kernels
// ---------------------------------------------------------------------------
__global__ void init_deg_kernel(float* __restrict__ deg, int n) {
    int i = blockIdx.x * blockDim.x + threadIdx.x;
    if (i < n) deg[i] = 1.0f;  // self-loop contributes 1 to every node's degree
}

__global__ void deg_accum_kernel(const long long* __restrict__ ei,
                                 float* __restrict__ deg, int E) {
    int e = blockIdx.x * blockDim.x + threadIdx.x;
    if (e < E) {
        int dst = (int)ei[(size_t)E + e];  // row 1 of edge_index
        atomicAdd(&deg[dst], 1.0f);
    }
}

__global__ void dinv_kernel(const float* __restrict__ deg,
                            float* __restrict__ dinv, int n) {
    int i = blockIdx.x * blockDim.x + threadIdx.x;
    if (i < n) dinv[i] = rsqrtf(deg[i]);  // deg >= 1 always (self-loops)
}

// ---------------------------------------------------------------------------
// GEMM: Out[M,DD] = (relu_in ? relu(X) : X) @ W^T (+ bias), via WMMA f32
// W is [out=DD, in=DD] row-major, so Out[m,o] = sum_k X[m,k] * W[o,k].
// One block = 8 waves = one 16-row strip; wave w computes 16x16 tile at n0=16w.
// ---------------------------------------------------------------------------
__global__ __launch_bounds__(256) void gemm_wmma_kernel(
    const float* __restrict__ X, const float* __restrict__ W,
    const float* __restrict__ bias, float* __restrict__ Out,
    int M, int relu_in) {
    __shared__ float sA[16 * DD];  // 8 KB: this block's 16 rows of X
    const int t  = threadIdx.x;
    const int m0 = blockIdx.x * 16;

    // Stage A block into LDS (512 float4 by 256 threads), fusing ReLU.
    const float4* X4  = (const float4*)(X + (size_t)m0 * DD);
    float4*       sA4 = (float4*)sA;
#pragma unroll
    for (int i = 0; i < 2; ++i) {
        float4 v = X4[t + i * 256];
        if (relu_in) {
            v.x = fmaxf(v.x, 0.0f); v.y = fmaxf(v.y, 0.0f);
            v.z = fmaxf(v.z, 0.0f); v.w = fmaxf(v.w, 0.0f);
        }
        sA4[t + i * 256] = v;
    }
    __syncthreads();

    const int lane = t & 31;
    const int r    = lane & 15;   // row-in-tile / col-in-tile index
    const int h    = lane >> 4;   // half-wave select (K pair / M+8)
    const int n0   = (t >> 5) * 16;

    v8f acc = {0.f, 0.f, 0.f, 0.f, 0.f, 0.f, 0.f, 0.f};
#pragma unroll
    for (int k = 0; k < DD; k += 4) {
        const int ka = k + 2 * h;
        v2f a, b;
        // A 16x4 frag: lanes 0-15 hold K={k,k+1}, lanes 16-31 hold K={k+2,k+3}
        a.x = sA[r * DD + ka];
        a.y = sA[r * DD + ka + 1];
        // B 4x16 frag = W^T tile: B[kk][n0+r] = W[n0+r][kk]
        b.x = W[(n0 + r) * DD + ka];
        b.y = W[(n0 + r) * DD + ka + 1];
        acc = __builtin_amdgcn_wmma_f32_16x16x4_f32(
            false, a, false, b, (short)0, acc, false, false);
    }

    // C/D layout: VGPR v -> row m0 + v + 8*h, col n0 + r
#pragma unroll
    for (int v = 0; v < 8; ++v) {
        const int row = m0 + v + 8 * h;
        const int col = n0 + r;
        float val = acc[v];
        if (bias) val += bias[col];
        Out[(size_t)row * DD + col] = val;
    }
}

// ---------------------------------------------------------------------------
// Scatter phase: Out = bias + selfloop + sum_edges norm * H[src]
// ---------------------------------------------------------------------------
__global__ void scatter_init_kernel(const float* __restrict__ H,
                                    const float* __restrict__ dinv,
                                    const float* __restrict__ bias,
                                    float* __restrict__ Out, int n) {
    int idx = blockIdx.x * blockDim.x + threadIdx.x;
    if (idx < n * DD) {
        int i = idx >> 7;          // node
        int c = idx & (DD - 1);    // channel
        float di = dinv[i];
        Out[idx] = bias[c] + H[idx] * di * di;  // self-loop norm = 1/deg
    }
}

__global__ void scatter_edge_kernel(const float* __restrict__ H,
                                    const long long* __restrict__ ei,
                                    const float* __restrict__ dinv,
                                    float* __restrict__ Out, int E) {
    // one wave32 per edge; each lane handles 4 channels (float4)
    int gt = blockIdx.x * blockDim.x + threadIdx.x;
    int e  = gt >> 5;
    if (e < E) {
        int src = (int)ei[e];
        int dst = (int)ei[(size_t)E + e];
        float norm = dinv[src] * dinv[dst];
        int c = (threadIdx.x & 31) * 4;
        const float4 hv = *(const float4*)(H + (size_t)src * DD + c);
        float* o = Out + (size_t)dst * DD + c;
        atomicAdd(o + 0, hv.x * norm);
        atomicAdd(o + 1, hv.y * norm);
        atomicAdd(o + 2, hv.z * norm);
        atomicAdd(o + 3, hv.w * norm);
    }
}

// ---------------------------------------------------------------------------
extern "C" void kernel_launch(void* const* d_in, const int* in_sizes, int n_in,
                              void* d_out, int out_size, void* d_ws, size_t ws_size,
                              hipStream_t stream) {
    const float*     x   = (const float*)d_in[0];
    const long long* ei  = (const long long*)d_in[1];  // int64 [2, E] flat
    const float*     W1  = (const float*)d_in[2];
    const float*     b1  = (const float*)d_in[3];
    const float*     W2  = (const float*)d_in[4];
    const float*     b2  = (const float*)d_in[5];
    const float*     Wfc = (const float*)d_in[6];
    const float*     bfc = (const float*)d_in[7];
    float*           out = (float*)d_out;

    const int N = in_sizes[0] / DD;   // 100000 (divisible by 16)
    const int E = in_sizes[1] / 2;    // 1600000

    // Workspace layout (512B-aligned slices): deg | dinv | bufA | bufB
    auto align512 = [](size_t v) { return (v + 511) & ~(size_t)511; };
    const size_t featBytes = (size_t)N * DD * sizeof(float);
    char*  ws   = (char*)d_ws;
    size_t off  = 0;
    float* deg  = (float*)(ws + off); off += align512((size_t)N * sizeof(float));
    float* dinv = (float*)(ws + off); off += align512((size_t)N * sizeof(float));
    float* bufA = (float*)(ws + off); off += align512(featBytes);
    float* bufB = (float*)(ws + off);

    const int TPB = 256;
    const int gN    = (N + TPB - 1) / TPB;
    const int gE    = (E + TPB - 1) / TPB;
    const int gFeat = (N * DD + TPB - 1) / TPB;
    const int gEw   = (int)(((size_t)E * 32 + TPB - 1) / TPB);
    const int gGemm = N / 16;

    // --- normalization ---
    init_deg_kernel<<<gN, TPB, 0, stream>>>(deg, N);
    deg_accum_kernel<<<gE, TPB, 0, stream>>>(ei, deg, E);
    dinv_kernel<<<gN, TPB, 0, stream>>>(deg, dinv, N);

    // --- layer 1: h = x @ W1^T ; aggregate ; (+b1) ---
    gemm_wmma_kernel<<<gGemm, TPB, 0, stream>>>(x, W1, nullptr, bufA, N, 0);
    scatter_init_kernel<<<gFeat, TPB, 0, stream>>>(bufA, dinv, b1, bufB, N);
    scatter_edge_kernel<<<gEw, TPB, 0, stream>>>(bufA, ei, dinv, bufB, E);

    // --- layer 2: h = relu(h1) @ W2^T ; aggregate ; (+b2) ---
    gemm_wmma_kernel<<<gGemm, TPB, 0, stream>>>(bufB, W2, nullptr, bufA, N, 1);
    scatter_init_kernel<<<gFeat, TPB, 0, stream>>>(bufA, dinv, b2, bufB, N);
    scatter_edge_kernel<<<gEw, TPB, 0, stream>>>(bufA, ei, dinv, bufB, E);

    // --- fc: out = relu(h2) @ Wfc^T + bfc ---
    gemm_wmma_kernel<<<gGemm, TPB, 0, stream>>>(bufB, Wfc, bfc, out, N, 1);
}